// UpDownResample_19885698581049
// MI455X (gfx1250) — compile-verified
//
#include <hip/hip_runtime.h>
#include <stdint.h>

typedef float v2f __attribute__((ext_vector_type(2)));
typedef float v8f __attribute__((ext_vector_type(8)));

#define T_IN   4096
#define T_OUT  2048
#define ROWS_PB 16       // rows per block tile
#define TOUT_PB 128      // outputs per block tile (16 per wave x 8 waves)
#define SEG_LEN 272      // staged input floats per row (j = 0..271)
#define LDS_STRIDE 276   // row stride in LDS floats (bank-conflict free, 16B aligned)

__device__ __forceinline__ int reflect_idx(int g) {
    g = (g < 0) ? -g : g;
    g = (g >= T_IN) ? (2 * (T_IN - 1) - g) : g;
    return g;
}

__global__ __launch_bounds__(256)
void fir_down2_wmma(const float* __restrict__ x,
                    const float* __restrict__ wk,
                    float* __restrict__ y) {
    __shared__ float lds[ROWS_PB * LDS_STRIDE];
    __shared__ float swk[8];

    const int tid   = threadIdx.x;
    const int t0    = blockIdx.x * TOUT_PB;        // first output index of tile
    const int r0    = blockIdx.y * ROWS_PB;        // first row of tile
    const int gbase = 2 * t0 - 8;                  // input index of LDS j=0 (mult of 4 elems)

    if (tid < 8) swk[tid] = wk[tid];

    // ---- Stage 16 rows x 272 floats into LDS.
    // Interior fast path: CDNA5 async direct global->LDS b128 copy (ASYNCcnt),
    // no VGPR round-trip. Edge path: scalar reflect copy (DScnt).
    for (int q = tid; q < ROWS_PB * (SEG_LEN / 4); q += 256) {
        int row = q / (SEG_LEN / 4);
        int jq  = (q % (SEG_LEN / 4)) * 4;
        int g   = gbase + jq;
        float* dst = &lds[row * LDS_STRIDE + jq];
        if (g >= 0 && g + 3 < T_IN) {
            // GVS mode: saddr (SGPR64 base) + 32-bit per-lane byte offset.
            // Low 32 bits of the generic LDS pointer == LDS byte address.
            const unsigned goff  = (unsigned)(((size_t)(r0 + row) * T_IN + g)
                                              * sizeof(float));
            const unsigned laddr = (unsigned)(uintptr_t)dst;
            asm volatile("global_load_async_to_lds_b128 %0, %1, %2"
                         :
                         : "v"(laddr), "v"(goff), "s"(x)
                         : "memory");
        } else {
            const float* src = x + (size_t)(r0 + row) * T_IN;
            #pragma unroll
            for (int e = 0; e < 4; ++e) dst[e] = src[reflect_idx(g + e)];
        }
    }
    // Fence this wave's async copies; barrier makes all LDS writes visible.
    asm volatile("s_wait_asynccnt 0" ::: "memory");
    __syncthreads();

    const int lane = tid & 31;
    const int w    = tid >> 5;       // wave id 0..7 -> outputs [t0+16w, t0+16w+16)
    const int nm   = lane & 15;      // N (row) for B/D ; M (output) for A
    const int hi   = lane >> 4;      // half-wave select
    const int kh   = hi << 1;        // K sub-offset within a chunk: 0 or 2

    // ---- Banded weight matrix A, chunked over K: A[m][u] = wk[u - 2m] if 0<=u-2m<8
    // Unconditional clamped LDS read + v_cndmask (avoids exec-masked load blocks).
    v2f Am[10];
    #pragma unroll
    for (int c = 0; c < 10; ++c) {
        int d0 = 4 * c + kh - 2 * nm;
        int d1 = d0 + 1;
        float w0 = swk[d0 & 7];
        float w1 = swk[d1 & 7];
        Am[c][0] = (d0 >= 0 && d0 < 8) ? w0 : 0.0f;
        Am[c][1] = (d1 >= 0 && d1 < 8) ? w1 : 0.0f;
    }

    // Wave w's 40-float segment starts at lds j = 32w + 5  (== input 2(t0+16w)-3)
    const int jw = 32 * w + 5;
    const float* seg = &lds[nm * LDS_STRIDE + jw + kh];

    v8f acc = {0.f, 0.f, 0.f, 0.f, 0.f, 0.f, 0.f, 0.f};
    #pragma unroll
    for (int c = 0; c < 10; ++c) {
        v2f b;
        b[0] = seg[4 * c];
        b[1] = seg[4 * c + 1];
        acc = __builtin_amdgcn_wmma_f32_16x16x4_f32(false, Am[c], false, b,
                                                    (short)0, acc, false, false);
    }

    // ---- D: lane holds 8 consecutive outputs of one row; two aligned b128 stores
    size_t obase = (size_t)(r0 + nm) * T_OUT + (size_t)(t0 + 16 * w + 8 * hi);
    *(float4*)(y + obase)     = make_float4(acc[0], acc[1], acc[2], acc[3]);
    *(float4*)(y + obase + 4) = make_float4(acc[4], acc[5], acc[6], acc[7]);
}

extern "C" void kernel_launch(void* const* d_in, const int* in_sizes, int n_in,
                              void* d_out, int out_size, void* d_ws, size_t ws_size,
                              hipStream_t stream) {
    const float* x  = (const float*)d_in[0];
    const float* wk = (const float*)d_in[1];
    float* y        = (float*)d_out;

    const int R = 4 * 64 * 128;                 // 32768 independent rows
    dim3 grid(T_OUT / TOUT_PB, R / ROWS_PB);    // (16, 2048)
    dim3 block(256);
    fir_down2_wmma<<<grid, block, 0, stream>>>(x, wk, y);
}